// SpectralConv1d_61443802137158
// MI455X (gfx1250) — compile-verified
//
#include <hip/hip_runtime.h>

typedef __attribute__((ext_vector_type(2))) float v2f;
typedef __attribute__((ext_vector_type(8))) float v8f;

#define BATCH   8
#define SEQ     2048
#define D_IN    32
#define D_OUT   32
#define KMAX    64
#define M_TOTAL (BATCH * SEQ)              // 16384 rows
#define ROW_STRIDE     (D_IN * KMAX * 2)   // 4096 floats per (b,s) row of x
#define OUT_ROW_STRIDE (D_OUT * KMAX * 2)  // 4096 floats per (b,s) row of out
#define W_O_STRIDE     (D_IN * KMAX * 2)   // 4096 floats per o of weight
#define I_STRIDE       (KMAX * 2)          // 128 floats per i

// Expanded-B LDS layout: per (i, o) store 4 floats {wr, -wi, wi, wr} so both
// WMMA B operands are direct even-aligned b64 LDS loads (no xor/mov in loop).
// i-stride of 130 words => odd-i lanes hit banks shifted by 2 (conflict-free).
#define LDS_I_STRIDE 130
#define LDS_FLOATS   (D_IN * LDS_I_STRIDE + 8)

// One block: 256 threads = 8 waves; handles 256 rows x 1 frequency mode k.
// One wave: 32 rows x 32 outputs x {re,im} for one k via 128 f32 WMMAs.
__global__ __launch_bounds__(256) void spectral_conv1d_wmma(
    const float* __restrict__ x, const float* __restrict__ w,
    float* __restrict__ out) {
  __shared__ __align__(16) float wlds[LDS_FLOATS];

  const int tid  = threadIdx.x;
  const int bid  = blockIdx.x;
  const int k    = bid & (KMAX - 1);   // frequency mode
  const int mblk = bid >> 6;           // 0..63, 256 rows each
  const int row0 = mblk * 256;

  // Stage this k's weight slice, pre-negated/swapped: {wr, -wi, wi, wr}.
  for (int p = tid; p < D_OUT * D_IN; p += 256) {
    const int o = p >> 5;
    const int i = p & 31;
    const float* src = w + (size_t)o * W_O_STRIDE + i * I_STRIDE + k * 2;
    const float wr = src[0];
    const float wi = src[1];
    const int idx = i * LDS_I_STRIDE + o * 4;
    *(v2f*)&wlds[idx]     = (v2f){wr, -wi};  // B_real rows  [wr; -wi]
    *(v2f*)&wlds[idx + 2] = (v2f){wi,  wr};  // B_imag rows  [wi;  wr]
  }
  __syncthreads();

  const int wave     = tid >> 5;
  const int lane     = tid & 31;
  const int lane_lo  = lane & 15;
  const int lane_hi  = lane >> 4;
  const int row_base = row0 + wave * 32;
  const int m0       = row_base + lane_lo;       // A-tile 0 row
  const int m1       = m0 + 16;                  // A-tile 1 row

  const float* xrow0 = x + (size_t)m0 * ROW_STRIDE + (size_t)k * 2;
  const float* xrow1 = x + (size_t)m1 * ROW_STRIDE + (size_t)k * 2;

  v8f c00r = {}, c00i = {};  // m-tile 0, o-tile 0 (o=0..15)
  v8f c01r = {}, c01i = {};  // m-tile 0, o-tile 1 (o=16..31)
  v8f c10r = {}, c10i = {};  // m-tile 1, o-tile 0
  v8f c11r = {}, c11i = {};  // m-tile 1, o-tile 1

#pragma unroll
  for (int s = 0; s < 16; ++s) {
    const int i = 2 * s + lane_hi;

    // A fragments (16x4 f32): K = {4s,4s+1} (low lanes) / {4s+2,4s+3} (high).
    const v2f a0 = __builtin_nontemporal_load((const v2f*)(xrow0 + i * I_STRIDE));
    const v2f a1 = __builtin_nontemporal_load((const v2f*)(xrow1 + i * I_STRIDE));

    // B fragments, pre-expanded in LDS: direct b64 loads, no fixup VALU.
    const int base = i * LDS_I_STRIDE + lane_lo * 4;
    const v2f br0 = *(const v2f*)&wlds[base];       // o-tile 0, real path
    const v2f bi0 = *(const v2f*)&wlds[base + 2];   // o-tile 0, imag path
    const v2f br1 = *(const v2f*)&wlds[base + 64];  // o-tile 1, real path
    const v2f bi1 = *(const v2f*)&wlds[base + 66];  // o-tile 1, imag path

    c00r = __builtin_amdgcn_wmma_f32_16x16x4_f32(false, a0, false, br0,
                                                 (short)0, c00r, false, false);
    c00i = __builtin_amdgcn_wmma_f32_16x16x4_f32(false, a0, false, bi0,
                                                 (short)0, c00i, false, false);
    c01r = __builtin_amdgcn_wmma_f32_16x16x4_f32(false, a0, false, br1,
                                                 (short)0, c01r, false, false);
    c01i = __builtin_amdgcn_wmma_f32_16x16x4_f32(false, a0, false, bi1,
                                                 (short)0, c01i, false, false);
    c10r = __builtin_amdgcn_wmma_f32_16x16x4_f32(false, a1, false, br0,
                                                 (short)0, c10r, false, false);
    c10i = __builtin_amdgcn_wmma_f32_16x16x4_f32(false, a1, false, bi0,
                                                 (short)0, c10i, false, false);
    c11r = __builtin_amdgcn_wmma_f32_16x16x4_f32(false, a1, false, br1,
                                                 (short)0, c11r, false, false);
    c11i = __builtin_amdgcn_wmma_f32_16x16x4_f32(false, a1, false, bi1,
                                                 (short)0, c11i, false, false);
  }

  // C/D layout: element (VGPR r, lane) -> M = r + 8*lane_hi, N = lane_lo.
  // Pack {re,im} accumulators into contiguous float2 NT stores (c stride 1).
  const int mo0 = row_base + 8 * lane_hi;        // m-tile 0 store base
  const int mo1 = mo0 + 16;                      // m-tile 1 store base
#pragma unroll
  for (int r = 0; r < 8; ++r) {
    const size_t off0 = (size_t)(mo0 + r) * OUT_ROW_STRIDE + (size_t)k * 2;
    const size_t off1 = (size_t)(mo1 + r) * OUT_ROW_STRIDE + (size_t)k * 2;
    __builtin_nontemporal_store((v2f){c00r[r], c00i[r]},
                                (v2f*)(out + off0 + lane_lo * I_STRIDE));
    __builtin_nontemporal_store((v2f){c01r[r], c01i[r]},
                                (v2f*)(out + off0 + (16 + lane_lo) * I_STRIDE));
    __builtin_nontemporal_store((v2f){c10r[r], c10i[r]},
                                (v2f*)(out + off1 + lane_lo * I_STRIDE));
    __builtin_nontemporal_store((v2f){c11r[r], c11i[r]},
                                (v2f*)(out + off1 + (16 + lane_lo) * I_STRIDE));
  }
}

extern "C" void kernel_launch(void* const* d_in, const int* in_sizes, int n_in,
                              void* d_out, int out_size, void* d_ws, size_t ws_size,
                              hipStream_t stream) {
  const float* x = (const float*)d_in[0];   // (8, 2048, 32, 64, 2) f32
  const float* w = (const float*)d_in[1];   // (32, 32, 64, 2) f32
  float* out = (float*)d_out;               // (8, 2048, 32, 64, 2) f32

  const int grid = (M_TOTAL / 256) * KMAX;  // 64 m-blocks * 64 modes = 4096
  spectral_conv1d_wmma<<<grid, 256, 0, stream>>>(x, w, out);
}